// IPLayer_12532714569874
// MI455X (gfx1250) — compile-verified
//
#include <hip/hip_runtime.h>

// segment_sum / scatter-add: out[idx[e], :] += inter[e, :]
// N_EDGES = 1,600,000, D_FEAT = 48, N_NODES = 50,000.
// Memory-bound: 307 MB streaming read + 76.8M f32 atomics into a 9.6 MB
// output that lives entirely in MI455X's 192 MB L2.

#define D_FEAT 48
#define VEC_PER_EDGE (D_FEAT / 4)   // 12 float4 per edge

// Native clang vector: accepted by __builtin_nontemporal_load; 16-byte,
// 16-aligned -> global_load_b128.
typedef float v4f __attribute__((ext_vector_type(4)));

// Guaranteed-native non-returning f32 atomic add (GLOBAL_ATOMIC_ADD_F32,
// ISA §15.18.3 op 86). Address in a VGPR pair, SADDR=off. No return value
// requested -> tracked on STOREcnt, no data comes back.
__device__ __forceinline__ void global_fadd(float* p, float v) {
    asm volatile("global_atomic_add_f32 %0, %1, off"
                 :
                 : "v"(p), "v"(v)
                 : "memory");
}

__global__ void seg_scatter_add_kernel(const int* __restrict__ idx,
                                       const float* __restrict__ inter,
                                       float* __restrict__ out,
                                       int n_edges) {
    int t = blockIdx.x * blockDim.x + threadIdx.x;
    int edge = t / VEC_PER_EDGE;          // magic-multiply division, cheap
    int q    = t - edge * VEC_PER_EDGE;   // which float4 of the 48-float row
    if (edge >= n_edges) return;

    // Streaming, read-once: non-temporal hint keeps the 307 MB stream from
    // evicting the hot 9.6 MB of output lines the L2 atomic units are using.
    const v4f* src = (const v4f*)(inter + (size_t)edge * D_FEAT) + q;
    v4f v = __builtin_nontemporal_load(src);

    int node = idx[edge];                 // 12 lanes share one cacheline: coalesced
    float* dst = out + (size_t)node * D_FEAT + q * 4;

    global_fadd(dst + 0, v.x);
    global_fadd(dst + 1, v.y);
    global_fadd(dst + 2, v.z);
    global_fadd(dst + 3, v.w);
}

__global__ void seg_zero_kernel(v4f* __restrict__ out, int n4) {
    int i = blockIdx.x * blockDim.x + threadIdx.x;
    if (i < n4) {
        out[i] = (v4f){0.f, 0.f, 0.f, 0.f};
    }
}

__global__ void seg_zero_tail_kernel(float* __restrict__ out, int n) {
    int i = blockIdx.x * blockDim.x + threadIdx.x;
    if (i < n) out[i] = 0.f;
}

extern "C" void kernel_launch(void* const* d_in, const int* in_sizes, int n_in,
                              void* d_out, int out_size, void* d_ws, size_t ws_size,
                              hipStream_t stream) {
    const int*   idx   = (const int*)d_in[0];    // idx_i: [N_EDGES] int32
    const float* inter = (const float*)d_in[1];  // inter: [N_EDGES, 48] f32
    float*       out   = (float*)d_out;          // [N_NODES, 48] f32

    const int n_edges = in_sizes[0];

    // 1) Zero the output (harness poisons d_out; graph replay re-accumulates).
    //    out_size = num_segments * 48 -> divisible by 4; tail path is
    //    defensive only (host-side constant branch, deterministic).
    int n4   = out_size / 4;
    int tail = out_size - n4 * 4;
    if (n4 > 0) {
        int threads = 256;
        int blocks  = (n4 + threads - 1) / threads;
        seg_zero_kernel<<<blocks, threads, 0, stream>>>((v4f*)out, n4);
    }
    if (tail > 0) {
        seg_zero_tail_kernel<<<1, 32, 0, stream>>>(out + (size_t)n4 * 4, tail);
    }

    // 2) Scatter-add: 12 threads per edge, one float4 load + 4 atomics each.
    {
        long long total   = (long long)n_edges * VEC_PER_EDGE;
        int       threads = 256;
        long long blocks  = (total + threads - 1) / threads;
        seg_scatter_add_kernel<<<(int)blocks, threads, 0, stream>>>(
            idx, inter, out, n_edges);
    }
}